// LatentLayer_40673340293239
// MI455X (gfx1250) — compile-verified
//
#include <hip/hip_runtime.h>

typedef float v2f __attribute__((ext_vector_type(2)));
typedef float v8f __attribute__((ext_vector_type(8)));

#define N_Z 8192
#define M_E 16384
#define EPS_N 1024
#define S_CHUNKS 8
#define ROWS_PER_CHUNK (N_Z / S_CHUNKS)        // 1024
#define TILES_PER_CHUNK (ROWS_PER_CHUNK / 16)  // 64
#define WAVES_PER_BLOCK 8
#define COLS_PER_BLOCK (WAVES_PER_BLOCK * 16)  // 128

#define LOG2E 1.4426950408889634f
#define LN2   0.6931471805599453f
#define LN_N  9.010913347279288f               // ln(8192) = 13*ln(2)

// Stage 1: each wave owns 16 e-columns; loop 16-row z tiles from LDS through
// V_WMMA_F32_16X16X4_F32 (K=2 padded to 4), exp2 each element, accumulate
// per-column partial sums of exp(alpha * d) for one N-chunk.
__global__ __launch_bounds__(256)
void kde_partial_kernel(const float* __restrict__ z,
                        const float* __restrict__ emb,
                        const float* __restrict__ log_sigma,
                        const float* __restrict__ eps,
                        const float* __restrict__ temperature,
                        float* __restrict__ partial) {
  __shared__ __align__(16) float sZ[2 * ROWS_PER_CHUNK + 2];  // (x,y) pairs + zero slot
  __shared__ __align__(16) float sZZ[ROWS_PER_CHUNK];         // cE * ||z||^2

  const float T  = temperature[0];
  const float ls = log_sigma[0];
  // alpha = -1/(2*sigma^2) = -0.5 * exp(-2*ls)
  const float alpha = -0.5f * __builtin_amdgcn_exp2f((-2.0f * ls) * LOG2E);
  const float cE    = alpha * LOG2E;   // so exp(alpha*x) == exp2(cE*x)
  const float m2cE  = -2.0f * cE;

  const int tid     = threadIdx.x;
  const int chunk   = blockIdx.y;
  const int rowBase = chunk * ROWS_PER_CHUNK;

  for (int i = tid; i < ROWS_PER_CHUNK; i += 256) {
    const float x = z[2 * (rowBase + i)];
    const float y = z[2 * (rowBase + i) + 1];
    sZ[2 * i]     = x;
    sZ[2 * i + 1] = y;
    sZZ[i]        = cE * (x * x + y * y);
  }
  if (tid == 0) {  // zero pair for the K=2,3 padding lanes of the A matrix
    sZ[2 * ROWS_PER_CHUNK]     = 0.0f;
    sZ[2 * ROWS_PER_CHUNK + 1] = 0.0f;
  }
  __syncthreads();

  const int  wave = tid >> 5;
  const int  lane = tid & 31;
  const bool lo   = (lane < 16);
  const int  col  = blockIdx.x * COLS_PER_BLOCK + (wave << 4) + (lane & 15);

  // e[col] = (1-T)*emb[col/1024] + T*eps[col%1024]
  const int   k  = col >> 10;
  const int   jj = col & (EPS_N - 1);
  const float ex = (1.0f - T) * emb[2 * k]     + T * eps[2 * jj];
  const float ey = (1.0f - T) * emb[2 * k + 1] + T * eps[2 * jj + 1];
  const float eeE = cE * (ex * ex + ey * ey);

  // B matrix 4x16: VGPR0 = row K=0 (lanes 0-15) / K=2 (lanes 16-31), VGPR1 = K=1/K=3.
  v2f b;
  b.x = lo ? ex : 0.0f;
  b.y = lo ? ey : 0.0f;

  const float2* sZ2 = reinterpret_cast<const float2*>(sZ);
  const int hi8 = (lane >> 4) << 3;  // C rows: +0 for lanes 0-15, +8 for 16-31
  float acc = 0.0f;

  for (int t = 0; t < TILES_PER_CHUNK; ++t) {
    const int r0 = t << 4;
    // A matrix 16x4: lanes 0-15 hold (z.x=K0, z.y=K1) for M=lane; lanes 16-31 hold K=2,3 = 0.
    const int    aIdx = lo ? (r0 + lane) : ROWS_PER_CHUNK;
    const float2 ap   = sZ2[aIdx];
    v2f a;
    a.x = ap.x;
    a.y = ap.y;

    const float4 q0 = *reinterpret_cast<const float4*>(&sZZ[r0 + hi8]);
    const float4 q1 = *reinterpret_cast<const float4*>(&sZZ[r0 + hi8 + 4]);

    v8f c = {};
    c = __builtin_amdgcn_wmma_f32_16x16x4_f32(
        /*neg_a=*/false, a, /*neg_b=*/false, b,
        /*c_mod=*/(short)0, c, /*reuse_a=*/false, /*reuse_b=*/false);

    // exp(alpha*d) = exp2(cE*zz + cE*ee - 2*cE*(z.e))
    acc += __builtin_amdgcn_exp2f(fmaf(c[0], m2cE, q0.x + eeE));
    acc += __builtin_amdgcn_exp2f(fmaf(c[1], m2cE, q0.y + eeE));
    acc += __builtin_amdgcn_exp2f(fmaf(c[2], m2cE, q0.z + eeE));
    acc += __builtin_amdgcn_exp2f(fmaf(c[3], m2cE, q0.w + eeE));
    acc += __builtin_amdgcn_exp2f(fmaf(c[4], m2cE, q1.x + eeE));
    acc += __builtin_amdgcn_exp2f(fmaf(c[5], m2cE, q1.y + eeE));
    acc += __builtin_amdgcn_exp2f(fmaf(c[6], m2cE, q1.z + eeE));
    acc += __builtin_amdgcn_exp2f(fmaf(c[7], m2cE, q1.w + eeE));
  }

  // Column j lives in lanes j and j+16 -> pairwise combine, lanes 0-15 write.
  const float other = __shfl_xor(acc, 16, 32);
  const float sum   = acc + other;
  if (lo) partial[col * S_CHUNKS + chunk] = sum;
}

// Stage 2: per-column chunk sums -> log -> mean -> scalar loss. Single block,
// fixed reduction order => deterministic across graph replays.
__global__ __launch_bounds__(1024)
void kde_final_kernel(const float* __restrict__ partial,
                      const float* __restrict__ log_sigma,
                      float* __restrict__ out) {
  __shared__ float red[1024];
  const int tid = threadIdx.x;
  float local = 0.0f;
  for (int j = tid; j < M_E; j += 1024) {
    float s = 0.0f;
#pragma unroll
    for (int c2 = 0; c2 < S_CHUNKS; ++c2) s += partial[j * S_CHUNKS + c2];
    local += LN2 * __builtin_amdgcn_logf(s);  // ln(s) = ln2 * log2(s)
  }
  red[tid] = local;
  __syncthreads();
  for (int off = 512; off > 0; off >>= 1) {
    if (tid < off) red[tid] += red[tid + off];
    __syncthreads();
  }
  if (tid == 0) {
    const float ls = log_sigma[0];
    // 0.5 * z_dim * (2*ls - 1) with z_dim == 2
    out[0] = -(red[0] / (float)M_E) + (2.0f * ls - 1.0f) + LN_N;
  }
}

extern "C" void kernel_launch(void* const* d_in, const int* in_sizes, int n_in,
                              void* d_out, int out_size, void* d_ws, size_t ws_size,
                              hipStream_t stream) {
  const float* z           = (const float*)d_in[0];
  const float* emb         = (const float*)d_in[1];
  const float* log_sigma   = (const float*)d_in[2];
  const float* eps         = (const float*)d_in[3];
  const float* temperature = (const float*)d_in[4];
  float* out     = (float*)d_out;
  float* partial = (float*)d_ws;  // M_E * S_CHUNKS floats = 512 KB

  dim3 grid(M_E / COLS_PER_BLOCK, S_CHUNKS);  // 128 x 8 blocks, 256 threads (8 waves)
  kde_partial_kernel<<<grid, 256, 0, stream>>>(z, emb, log_sigma, eps, temperature, partial);
  kde_final_kernel<<<1, 1024, 0, stream>>>(partial, log_sigma, out);
}